// get_loss_7834020348217
// MI455X (gfx1250) — compile-verified
//
#include <hip/hip_runtime.h>

typedef __attribute__((ext_vector_type(2))) float v2f;
typedef __attribute__((ext_vector_type(8))) float v8f;

#define WAVES_PER_BLOCK 8

// One wave computes min_j d(p_i, q_j) for a 16-row tile of P against all of Q,
// using V_WMMA_F32_16X16X4_F32 with the |p|^2 term folded into the K=3 slot:
//   A fragment: 16x4 f32, A[m,*] = (-2px, -2py, -2pz, |p_m|^2)
//   B fragment: 4x16 f32, B[*,n] = ( qx,   qy,   qz,   1     )
//   D[m,n] = |p_m|^2 - 2*dot(p_m, q_n);   d = D + |q_n|^2
__global__ __launch_bounds__(256) void chamfer_rowmin_wmma(
    const float* __restrict__ P, const float* __restrict__ Q,
    float* __restrict__ rowmin_out, int Nq, int Nt, int Bsz)
{
    const int lane = threadIdx.x & 31;
    const int wave = threadIdx.x >> 5;
    const int half = lane >> 4;        // 0: K=0,1 / rows v ; 1: K=2,3 / rows v+8
    const int lidx = lane & 15;

    const int tilesPerBatch = Nq >> 4;
    const int g     = blockIdx.x * WAVES_PER_BLOCK + wave;
    const int batch = g / tilesPerBatch;
    const int tile  = g - batch * tilesPerBatch;
    if (batch >= Bsz) return;          // wave-uniform; grid is exact anyway

    // ---- Load A tile (held in registers for the whole loop) ----
    const int row = tile * 16 + lidx;
    const float* prow = P + ((size_t)batch * Nq + row) * 3;
    const float px = prow[0], py = prow[1], pz = prow[2];
    const float x2 = px * px + py * py + pz * pz;

    v2f a;
    a.x = half ? (-2.0f * pz) : (-2.0f * px);   // K=2 : K=0
    a.y = half ? x2           : (-2.0f * py);   // K=3 : K=1

    float rm[8];
#pragma unroll
    for (int v = 0; v < 8; ++v) rm[v] = 3.402823466e+38f;

    const float* qbase  = Q + (size_t)batch * Nt * 3;
    const int    ctiles = Nt >> 4;

#pragma unroll 2
    for (int ct = 0; ct < ctiles; ++ct) {
        // prefetch a future B tile into cache (speculative)
        const int pf = (ct + 8 < ctiles) ? (ct + 8) : ct;
        __builtin_prefetch(qbase + ((size_t)(pf * 16 + lidx)) * 3, 0, 1);

        const float* qcol = qbase + ((size_t)(ct * 16 + lidx)) * 3;
        const float qx = qcol[0], qy = qcol[1], qz = qcol[2];
        const float y2 = qx * qx + qy * qy + qz * qz;   // col = lidx for this lane

        v2f bfrag;
        bfrag.x = half ? qz   : qx;    // K=2 : K=0
        bfrag.y = half ? 1.0f : qy;    // K=3 : K=1

        v8f c = {};
        // 8 args: (neg_a, A, neg_b, B, c_mod, C, reuse_a, reuse_b)
        c = __builtin_amdgcn_wmma_f32_16x16x4_f32(
                false, a, false, bfrag, (short)0, c, false, false);

#pragma unroll
        for (int v = 0; v < 8; ++v) {
            rm[v] = fminf(rm[v], c[v] + y2);
        }
    }

    // ---- reduce mins across the 16 lanes of each half-wave ----
#pragma unroll
    for (int v = 0; v < 8; ++v) {
        float m = rm[v];
        m = fminf(m, __shfl_xor(m, 1, 32));
        m = fminf(m, __shfl_xor(m, 2, 32));
        m = fminf(m, __shfl_xor(m, 4, 32));
        m = fminf(m, __shfl_xor(m, 8, 32));
        rm[v] = m;
    }

    if (lidx == 0) {                    // lanes 0 and 16 write their 8 rows
        const int rbase = batch * Nq + tile * 16 + half * 8;
#pragma unroll
        for (int v = 0; v < 8; ++v) rowmin_out[rbase + v] = rm[v];
    }
}

// Single-block final reduction: mean(first n1) + mean(next n2) + lam*fbpp
__global__ __launch_bounds__(512) void chamfer_finalize(
    const float* __restrict__ mins, const float* __restrict__ fbpp,
    const float* __restrict__ lam, float* __restrict__ out, int n1, int n2)
{
    __shared__ float s1[512];
    __shared__ float s2[512];
    const int tid = threadIdx.x;

    float a1 = 0.0f, a2 = 0.0f;
    for (int i = tid; i < n1; i += 512) a1 += mins[i];
    for (int i = tid; i < n2; i += 512) a2 += mins[n1 + i];
    s1[tid] = a1;
    s2[tid] = a2;
    __syncthreads();

    for (int s = 256; s > 0; s >>= 1) {
        if (tid < s) {
            s1[tid] += s1[tid + s];
            s2[tid] += s2[tid + s];
        }
        __syncthreads();
    }
    if (tid == 0)
        out[0] = s1[0] / (float)n1 + s2[0] / (float)n2 + lam[0] * fbpp[0];
}

extern "C" void kernel_launch(void* const* d_in, const int* in_sizes, int n_in,
                              void* d_out, int out_size, void* d_ws, size_t ws_size,
                              hipStream_t stream) {
    const float* pred = (const float*)d_in[0];   // (B,N,3) f32
    const float* targ = (const float*)d_in[1];   // (B,M,3) f32
    const float* fbpp = (const float*)d_in[2];   // scalar f32
    const float* lam  = (const float*)d_in[3];   // scalar f32
    float* out = (float*)d_out;
    float* ws  = (float*)d_ws;                   // B*N + B*M floats of row-mins

    const int B = 4;
    const int N = in_sizes[0] / (B * 3);
    const int M = in_sizes[1] / (B * 3);

    // direction 1: pred -> nearest in target (B*N row-mins)
    {
        const int waves = B * (N >> 4);
        chamfer_rowmin_wmma<<<waves / WAVES_PER_BLOCK, 256, 0, stream>>>(
            pred, targ, ws, N, M, B);
    }
    // direction 2: target -> nearest in pred (B*M row-mins)
    {
        const int waves = B * (M >> 4);
        chamfer_rowmin_wmma<<<waves / WAVES_PER_BLOCK, 256, 0, stream>>>(
            targ, pred, ws + (size_t)B * N, M, N, B);
    }
    chamfer_finalize<<<1, 512, 0, stream>>>(ws, fbpp, lam, out, B * N, B * M);
}